// MPSLinear_85529978732864
// MI455X (gfx1250) — compile-verified
//
#include <hip/hip_runtime.h>

// ---------------------------------------------------------------------------
// MPS linear forward for MI455X (gfx1250), wave32, fp32 WMMA path.
// One block per batch row; 8 waves: 4 left + 4 right chain segments of 98
// matrices. Chain runs as P <- M_w @ P with P held entirely in registers as
// 2x2 WMMA D-tiles (ping-pong, no copies); M_w (A operand) streams from
// global/L2 as b128 loads; P is recycled as the B operand with a single
// lane-half-swap per fragment word (v_permlane16_swap_b32 when available,
// else branchless ds_swizzle SWAPX16 + cndmask). Zero LDS in the hot loop.
// ---------------------------------------------------------------------------

namespace {

constexpr int VBD  = 20;    // bond dimension
constexpr int LS   = 34;    // LDS row stride (floats): even (b64 align) & conflict-free
constexpr int HN   = 392;
constexpr int SEG  = 98;    // 392 / 4 waves per side
constexpr int INF  = 784;
constexpr int OUTF = 10;
constexpr int MELE = VBD * VBD * 2;   // floats per weight matrix (20*20*2)

typedef __attribute__((ext_vector_type(2))) float    v2f;
typedef __attribute__((ext_vector_type(8))) float    v8f;
typedef __attribute__((ext_vector_type(2))) unsigned uv2;

__device__ __forceinline__ v8f wmma4(v2f a, v2f b, v8f c) {
  // D = A(16x4 f32) * B(4x16 f32) + C(16x16 f32)
  return __builtin_amdgcn_wmma_f32_16x16x4_f32(false, a, false, b,
                                               (short)0, c, false, false);
}

// merge_lo(qa,qb) = {qa on lanes 0-15, qb's lanes 0-15 on lanes 16-31}
// merge_hi(qa,qb) = {qa's lanes 16-31 on lanes 0-15, qb on lanes 16-31}
#if defined(__HIP_DEVICE_COMPILE__) && __has_builtin(__builtin_amdgcn_permlane16_swap)

__device__ __forceinline__ float merge_lo(float qa, float qb, bool) {
  uv2 r = __builtin_amdgcn_permlane16_swap(__float_as_uint(qa),
                                           __float_as_uint(qb), false, false);
  return __uint_as_float(r.x);     // {qa.lo, qb.lo}
}
__device__ __forceinline__ float merge_hi(float qa, float qb, bool) {
  uv2 r = __builtin_amdgcn_permlane16_swap(__float_as_uint(qa),
                                           __float_as_uint(qb), false, false);
  return __uint_as_float(r.y);     // {qa.hi, qb.hi}
}

#elif defined(__HIP_DEVICE_COMPILE__) && __has_builtin(__builtin_amdgcn_ds_swizzle)

// ds_swizzle_b32 group-of-32 SWAPX16: offset = xor(0x10)<<10 | and(0x1f)
__device__ __forceinline__ float swap16(float x) {
  return __uint_as_float(__builtin_amdgcn_ds_swizzle(__float_as_uint(x), 0x401F));
}
__device__ __forceinline__ float merge_lo(float qa, float qb, bool halfb) {
  const float s = swap16(qb);      // unconditional: keeps EXEC intact
  return halfb ? s : qa;
}
__device__ __forceinline__ float merge_hi(float qa, float qb, bool halfb) {
  const float s = swap16(qa);
  return halfb ? qb : s;
}

#else  // host-parse / generic fallback

__device__ __forceinline__ float merge_lo(float qa, float qb, bool halfb) {
  const float s = __shfl_xor(qb, 16, 32);
  return halfb ? s : qa;
}
__device__ __forceinline__ float merge_hi(float qa, float qb, bool halfb) {
  const float s = __shfl_xor(qa, 16, 32);
  return halfb ? qb : s;
}

#endif

// A fragment straight from the global weight tensor (rows per lane, K = p-dim
// neighbors -> 4 contiguous floats). Branchless row padding via cndmask.
__device__ __forceinline__ v2f a_frag_global(const float* tw, float x0, float x1,
                                             int mt, int kc, int l16, int half) {
  const int row  = 16 * mt + l16;
  const bool ok  = (row < VBD);            // compile-time true for mt==0
  const int rowc = ok ? row : 0;
  const int k0   = 4 * kc + 2 * half;      // k0,k0+1 < 20 always (kc<5)
  const float* e = tw + ((rowc * VBD + k0) << 1);
  const float ax = x0 * e[0] + x1 * e[1];
  const float ay = x0 * e[2] + x1 * e[3];
  v2f a;
  a.x = ok ? ax : 0.0f;
  a.y = ok ? ay : 0.0f;
  return a;
}

// Per-kc source location of the 4 needed P rows inside the D tiles:
//   rows 4kc..4kc+3 live in tile KMT, elements KV0..KV0+3, lane-half KH.
constexpr int KMT[5] = {0, 0, 0, 0, 1};
constexpr int KV0[5] = {0, 4, 0, 4, 0};
constexpr int KH [5] = {0, 0, 1, 1, 0};

// Rebuild a B fragment (4x16 slice of P) from the register D tiles.
// B VGPR v wants row 4kc+v+2*(lane>=16); only a half-swap is needed because
// the lane%16 (column) mapping of B and D layouts coincide.
__device__ __forceinline__ v2f b_frag_tiles(const v8f C[2][2], int kc, int nt,
                                            bool halfb) {
  const v8f& c = C[KMT[kc]][nt];
  const float q0 = c[KV0[kc] + 0];
  const float q1 = c[KV0[kc] + 1];
  const float q2 = c[KV0[kc] + 2];
  const float q3 = c[KV0[kc] + 3];
  v2f b;
  if (KH[kc] == 0) { b.x = merge_lo(q0, q2, halfb); b.y = merge_lo(q1, q3, halfb); }
  else             { b.x = merge_hi(q0, q2, halfb); b.y = merge_hi(q1, q3, halfb); }
  return b;
}

// One chain step: Cout = M_w @ Cin (all tiles in registers).
__device__ __forceinline__ void chain_step(const float* tens, int w, int w0,
                                           const float* input, size_t xbase,
                                           const v8f Cin[2][2], v8f Cout[2][2],
                                           int l16, int half, bool halfb) {
  const float* tw = tens + (size_t)w * MELE;
  float x0 = input[xbase + 2 * w + 0];
  float x1 = input[xbase + 2 * w + 1];
  x0 = __uint_as_float(__builtin_amdgcn_readfirstlane(__float_as_uint(x0)));
  x1 = __uint_as_float(__builtin_amdgcn_readfirstlane(__float_as_uint(x1)));

  if (w - 2 >= w0) {  // pull a later chain matrix (3.2 KB) toward L0
    const char* nxt = (const char*)(tens + (size_t)(w - 2) * MELE);
    __builtin_prefetch(nxt + (l16 + 16 * half) * 128, 0, 3);
  }

  const v8f vz = {0, 0, 0, 0, 0, 0, 0, 0};
  v8f n00 = vz, n01 = vz, n10 = vz, n11 = vz;
#pragma unroll
  for (int kc = 0; kc < 5; ++kc) {
    const v2f a0 = a_frag_global(tw, x0, x1, 0, kc, l16, half);
    const v2f a1 = a_frag_global(tw, x0, x1, 1, kc, l16, half);
    const v2f b0 = b_frag_tiles(Cin, kc, 0, halfb);
    const v2f b1 = b_frag_tiles(Cin, kc, 1, halfb);
    n00 = wmma4(a0, b0, n00);
    n01 = wmma4(a0, b1, n01);
    n10 = wmma4(a1, b0, n10);
    n11 = wmma4(a1, b1, n11);
  }
  Cout[0][0] = n00; Cout[0][1] = n01; Cout[1][0] = n10; Cout[1][1] = n11;
}

// 20x20 @ 20x20 with operands/dest in zero-padded LDS slabs (tree combine).
// All accesses unconditional: pad rows/cols hold exact zeros by invariant.
__device__ __forceinline__ void mm_lds(const float* A, const float* B, float* D,
                                       int l16, int half) {
  const v8f vz = {0, 0, 0, 0, 0, 0, 0, 0};
  v8f c00 = vz, c01 = vz, c10 = vz, c11 = vz;
#pragma unroll
  for (int kc = 0; kc < 5; ++kc) {
    const int k0 = 4 * kc + 2 * half;
    v2f a0, a1, b0, b1;
    a0.x = A[(0  + l16) * LS + k0];     a0.y = A[(0  + l16) * LS + k0 + 1];
    a1.x = A[(16 + l16) * LS + k0];     a1.y = A[(16 + l16) * LS + k0 + 1];
    b0.x = B[k0 * LS + l16];            b0.y = B[(k0 + 1) * LS + l16];
    b1.x = B[k0 * LS + 16 + l16];       b1.y = B[(k0 + 1) * LS + 16 + l16];
    c00 = wmma4(a0, b0, c00);
    c01 = wmma4(a0, b1, c01);
    c10 = wmma4(a1, b0, c10);
    c11 = wmma4(a1, b1, c11);
  }
  const v8f cc[2][2] = {{c00, c01}, {c10, c11}};
#pragma unroll
  for (int mt = 0; mt < 2; ++mt)
#pragma unroll
    for (int nt = 0; nt < 2; ++nt)
#pragma unroll
      for (int r = 0; r < 8; ++r)
        D[(16 * mt + r + 8 * half) * LS + 16 * nt + l16] = cc[mt][nt][r];
}

} // namespace

extern "C" __global__ __launch_bounds__(256)
void mps_forward_kernel(const float* __restrict__ input,  // (512,784,2)
                        const float* __restrict__ Tl,     // (392,20,20,2)
                        const float* __restrict__ Tr,     // (392,20,20,2)
                        const float* __restrict__ Tc,     // (10,20,20,1)
                        float* __restrict__ out) {        // (512,10,1)
  // 8 segment slabs + 4 level-1 scratch slabs, each 32 x LS, zero-padded.
  __shared__ float buf[12][32 * LS];               // 12 * 4352 B = 52.2 KB
  __shared__ float part[OUTF * VBD];

  const int b     = blockIdx.x;
  const int tid   = threadIdx.x;
  const int wave  = tid >> 5;
  const int lane  = tid & 31;
  const int l16   = lane & 15;
  const int half  = lane >> 4;
  const bool halfb = (half != 0);

  const int side = wave >> 2;             // 0 = left chain, 1 = right chain
  const int sw   = wave & 3;              // segment within side
  const float* tens = side ? Tr : Tl;
  const int w0 = sw * SEG;
  const size_t xbase = ((size_t)b * INF + (side ? HN : 0)) * 2;

  const v8f vz = {0, 0, 0, 0, 0, 0, 0, 0};
  v8f Ca[2][2], Cb[2][2];

  // ---- P = M_{w0+SEG-1}, loaded directly into D-tile register layout -----
  {
    const int wl = w0 + SEG - 1;
    const float* tw = tens + (size_t)wl * MELE;
    float x0 = input[xbase + 2 * wl + 0];
    float x1 = input[xbase + 2 * wl + 1];
    x0 = __uint_as_float(__builtin_amdgcn_readfirstlane(__float_as_uint(x0)));
    x1 = __uint_as_float(__builtin_amdgcn_readfirstlane(__float_as_uint(x1)));
#pragma unroll
    for (int mt = 0; mt < 2; ++mt)
#pragma unroll
      for (int nt = 0; nt < 2; ++nt) {
        v8f t = vz;
#pragma unroll
        for (int r = 0; r < 8; ++r) {
          const int row = 16 * mt + r + 8 * half;
          const int col = 16 * nt + l16;
          const bool ok = (row < VBD) && (col < VBD);
          const int off = ok ? ((row * VBD + col) << 1) : 0;
          const float v = x0 * tw[off] + x1 * tw[off + 1];
          t[r] = ok ? v : 0.0f;
        }
        Ca[mt][nt] = t;
      }
  }

  // ---- chain (reversed): P <- M_w @ P, ping-pong registers ---------------
  // 97 steps: 48 pairs (t = 96..1) + 1 tail (t = 0); result lands in Cb.
  for (int t = SEG - 2; t >= 1; t -= 2) {
    chain_step(tens, w0 + t,     w0, input, xbase, Ca, Cb, l16, half, halfb);
    chain_step(tens, w0 + t - 1, w0, input, xbase, Cb, Ca, l16, half, halfb);
  }
  chain_step(tens, w0, w0, input, xbase, Ca, Cb, l16, half, halfb);

  // ---- dump register P into this wave's LDS slab (pads stay zero) --------
  {
    float* Pb = buf[wave];
#pragma unroll
    for (int mt = 0; mt < 2; ++mt)
#pragma unroll
      for (int nt = 0; nt < 2; ++nt)
#pragma unroll
        for (int r = 0; r < 8; ++r)
          Pb[(16 * mt + r + 8 * half) * LS + 16 * nt + l16] = Cb[mt][nt][r];
  }
  __syncthreads();

  // ---- log-tree combine (scalar-uniform branches keep EXEC all-ones) -----
  const int swS   = __builtin_amdgcn_readfirstlane(sw);
  const int sideS = __builtin_amdgcn_readfirstlane(side);
  if (swS < 2) {   // level 1: (P0@P1)->scratch, (P2@P3)->scratch
    mm_lds(buf[sideS * 4 + 2 * swS], buf[sideS * 4 + 2 * swS + 1],
           buf[8 + sideS * 2 + swS], l16, half);
  }
  __syncthreads();
  if (swS == 0) {  // level 2: full side product -> buf[0] (L) / buf[4] (R)
    mm_lds(buf[8 + sideS * 2], buf[8 + sideS * 2 + 1],
           buf[sideS * 4], l16, half);
  }
  __syncthreads();

  const float* L = buf[0];
  const float* R = buf[4];

  // ---- out[b,o] = sum_{i,l} (sum_p L[i,p]*Tc[o,p,l]) * R[l,i] ------------
  if (tid < OUTF * VBD) {
    const int o = tid / VBD, i = tid % VBD;
    float acc = 0.0f;
    for (int l = 0; l < VBD; ++l) {
      float s = 0.0f;
#pragma unroll
      for (int p = 0; p < VBD; ++p)
        s += L[i * LS + p] * Tc[(o * VBD + p) * VBD + l];
      acc += s * R[l * LS + i];
    }
    part[tid] = acc;
  }
  __syncthreads();
  if (tid < OUTF) {
    float s = 0.0f;
#pragma unroll
    for (int i = 0; i < VBD; ++i) s += part[tid * VBD + i];
    out[(size_t)b * OUTF + tid] = s;
  }
}

extern "C" void kernel_launch(void* const* d_in, const int* in_sizes, int n_in,
                              void* d_out, int out_size, void* d_ws, size_t ws_size,
                              hipStream_t stream) {
  (void)in_sizes; (void)n_in; (void)out_size; (void)d_ws; (void)ws_size;
  const float* input = (const float*)d_in[0];   // (512,784,2)
  const float* Tl    = (const float*)d_in[1];   // (392,20,20,2)
  const float* Tr    = (const float*)d_in[2];   // (392,20,20,2)
  const float* Tc    = (const float*)d_in[3];   // (10,20,20,1)
  float* out         = (float*)d_out;           // (512,10,1)

  mps_forward_kernel<<<dim3(512), dim3(256), 0, stream>>>(input, Tl, Tr, Tc, out);
}